// CAM_Module_90211493085854
// MI455X (gfx1250) — compile-verified
//
#include <hip/hip_runtime.h>

typedef __attribute__((ext_vector_type(2))) float v2f;
typedef __attribute__((ext_vector_type(8))) float v8f;

#define B_      128
#define C_      1024
#define N_      49        // H*W = 7*7
#define NPAD    64        // 49 padded to 4 WMMA tiles
#define KCHUNKS 4
#define KCHUNK  (C_ / KCHUNKS)   // 256 channels per chunk
#define KSTEPS2 13        // ceil(49/4) K-steps needed in stage 2 (K=52 >= 49)

// ---------------------------------------------------------------------------
// Stage 1: Mp[chunk][b] (64x64 fp32, zero-padded) = k[b]^T @ v[b] over a
// 256-channel slice. One 32-thread block (single wave) per (batch, chunk).
// The wave owns the full 4x4 tile grid: 16 f32 accumulators (128 VGPRs),
// so each K-step is 16 loads : 16 WMMAs (1:1 VMEM:WMMA) with 16 independent
// accumulation chains. A-loads for the 4 i-tiles span one contiguous 49-float
// row of k (coalesces to a few cache lines); same for B over v.
// ---------------------------------------------------------------------------
__global__ __launch_bounds__(32) void cam_stage1(
    const float* __restrict__ kmat, const float* __restrict__ vmat,
    float* __restrict__ Mp)
{
    const int b     = blockIdx.x;
    const int chunk = blockIdx.y;
    const int lane  = threadIdx.x & 31;
    const int ml    = lane & 15;
    const int kh    = (lane >> 4) << 1;     // 0 for lanes 0-15, 2 for 16-31

    const float* kb = kmat + (size_t)b * C_ * N_;
    const float* vb = vmat + (size_t)b * C_ * N_;

    // Tile-local position masks (shared by A rows and B cols): pad >=49 -> 0.
    float sel[4];
    int   pc[4];
#pragma unroll
    for (int t = 0; t < 4; ++t) {
        int p  = t * 16 + ml;
        sel[t] = (p < N_) ? 1.0f : 0.0f;    // compile-time 1.0 for t < 3
        pc[t]  = (p < N_) ? p : (N_ - 1);   // clamped: always in-bounds
    }

    v8f acc[4][4] = {};                     // [i-tile][j-tile]

    const int c0 = chunk * KCHUNK;
    for (int c = c0; c < c0 + KCHUNK; c += 4) {
        const float* kr0 = kb + (size_t)(c + kh)     * N_;
        const float* kr1 = kb + (size_t)(c + kh + 1) * N_;
        const float* vr0 = vb + (size_t)(c + kh)     * N_;
        const float* vr1 = vb + (size_t)(c + kh + 1) * N_;

        v2f a[4], bm[4];
#pragma unroll
        for (int i = 0; i < 4; ++i) {       // A = k^T : VGPR0=K(kh), VGPR1=K(kh+1)
            a[i].x = kr0[pc[i]] * sel[i];
            a[i].y = kr1[pc[i]] * sel[i];
        }
#pragma unroll
        for (int j = 0; j < 4; ++j) {       // B = v : column per lane, same K split
            bm[j].x = vr0[pc[j]] * sel[j];
            bm[j].y = vr1[pc[j]] * sel[j];
        }
#pragma unroll
        for (int i = 0; i < 4; ++i)
#pragma unroll
            for (int j = 0; j < 4; ++j)
                acc[i][j] = __builtin_amdgcn_wmma_f32_16x16x4_f32(
                    false, a[i], false, bm[j], (short)0, acc[i][j], false, false);
    }

    // D layout: lane l, VGPR r -> row = r + 8*(l>=16), col = l&15
    float*    out  = Mp + (size_t)(chunk * B_ + b) * NPAD * NPAD;
    const int rhi  = (lane >> 4) << 3;
#pragma unroll
    for (int i = 0; i < 4; ++i)
#pragma unroll
        for (int j = 0; j < 4; ++j)
#pragma unroll
            for (int r = 0; r < 8; ++r)
                out[(size_t)(i * 16 + rhi + r) * NPAD + j * 16 + ml] = acc[i][j][r];
}

// ---------------------------------------------------------------------------
// Stage 2: out[b] = q[b] (1024x49) @ M[b] (49x49) + v1[b].
// One block per batch (8 waves). Reduce the 4 K-chunk partials into LDS,
// preload B into registers, sweep 8 row-tiles per wave.
// ---------------------------------------------------------------------------
__global__ __launch_bounds__(256) void cam_stage2(
    const float* __restrict__ q, const float* __restrict__ v1,
    const float* __restrict__ Mp, float* __restrict__ out)
{
    __shared__ float Ms[NPAD * NPAD];       // 16 KB
    const int b = blockIdx.x;

    for (int p = threadIdx.x; p < NPAD * NPAD; p += 256) {
        float s = 0.0f;
#pragma unroll
        for (int ch = 0; ch < KCHUNKS; ++ch)
            s += Mp[(size_t)(ch * B_ + b) * NPAD * NPAD + p];
        Ms[p] = s;
    }
    __syncthreads();

    const int wave = threadIdx.x >> 5;
    const int lane = threadIdx.x & 31;
    const int ml   = lane & 15;
    const int kh   = (lane >> 4) << 1;

    // Preload B = Ms: Breg[ks][j] covers K = 4*ks..4*ks+3, cols j*16..j*16+15.
    // Rows >= 49 of Ms are exact zeros (stage-1 padding), so K-pad is free.
    v2f Breg[KSTEPS2][4];
#pragma unroll
    for (int ks = 0; ks < KSTEPS2; ++ks)
#pragma unroll
        for (int j = 0; j < 4; ++j) {
            Breg[ks][j].x = Ms[(ks * 4 + kh)     * NPAD + j * 16 + ml];
            Breg[ks][j].y = Ms[(ks * 4 + kh + 1) * NPAD + j * 16 + ml];
        }

    const float* qb = q  + (size_t)b * C_ * N_;
    const float* vb = v1 + (size_t)b * C_ * N_;
    float*       ob = out + (size_t)b * C_ * N_;

    for (int it = wave; it < 64; it += 8) {          // 8 row-tiles per wave
        const int    mrow = it * 16 + ml;            // < 1024 always
        const float* qrow = qb + (size_t)mrow * N_;
        v8f acc[4] = {};
#pragma unroll
        for (int ks = 0; ks < KSTEPS2; ++ks) {
            const int kk0 = ks * 4 + kh;
            v2f a;                                   // contiguous -> b64 load
            a.x = (kk0     < N_) ? qrow[kk0]     : 0.0f;
            a.y = (kk0 + 1 < N_) ? qrow[kk0 + 1] : 0.0f;
#pragma unroll
            for (int j = 0; j < 4; ++j)
                acc[j] = __builtin_amdgcn_wmma_f32_16x16x4_f32(
                    false, a, false, Breg[ks][j], (short)0, acc[j], false, false);
        }
        // Epilogue: add v1 and store the valid 49 columns.
        const int rbase = it * 16 + ((lane >> 4) << 3);
#pragma unroll
        for (int j = 0; j < 4; ++j) {
            const int n = j * 16 + ml;
            if (n < N_) {
#pragma unroll
                for (int r = 0; r < 8; ++r) {
                    const size_t idx = (size_t)(rbase + r) * N_ + n;
                    ob[idx] = acc[j][r] + vb[idx];
                }
            }
        }
    }
}

// ---------------------------------------------------------------------------
extern "C" void kernel_launch(void* const* d_in, const int* in_sizes, int n_in,
                              void* d_out, int out_size, void* d_ws, size_t ws_size,
                              hipStream_t stream) {
    // setup_inputs order: v1, q1, k1  (all fp32)
    const float* v1 = (const float*)d_in[0];
    const float* q1 = (const float*)d_in[1];
    const float* k1 = (const float*)d_in[2];
    float*       out = (float*)d_out;
    float*       Mp  = (float*)d_ws;   // KCHUNKS * 128 * 64*64 floats = 8 MB

    dim3 g1(B_, KCHUNKS);
    cam_stage1<<<g1, dim3(32), 0, stream>>>(k1, v1, Mp);
    cam_stage2<<<dim3(B_), dim3(256), 0, stream>>>(q1, v1, Mp, out);
}